// MoleculeAttnBias_35631048688000
// MI455X (gfx1250) — compile-verified
//
#include <hip/hip_runtime.h>
#include <hip/hip_bf16.h>

// MoleculeAttnBias for MI455X (gfx1250, wave32).
// Wc = fold(W_edge, W3) precomputed in WMMA B-fragment order; main kernel is an
// fp32 WMMA GEMM (M=131072, K=1280, N=32) streaming edge_input (671 MB) into
// double-buffered LDS via GLOBAL_LOAD_ASYNC_TO_LDS_B128 (ASYNCcnt-tracked).
// 2Mx2N register blocking per wave: each A element is read from LDS exactly
// once; B fragments come straight from L2-resident global.

typedef __attribute__((ext_vector_type(2))) float v2f;
typedef __attribute__((ext_vector_type(8))) float v8f;

// Types matching the async-LDS builtin signature:
//   void __builtin_amdgcn_global_load_async_to_lds_b128(v4i AS1*, v4i AS3*, imm, imm)
typedef int v4i_ __attribute__((vector_size(16)));
typedef __attribute__((address_space(1))) v4i_ as1_v4i;
typedef __attribute__((address_space(3))) v4i_ as3_v4i;

#define NB   32          // batch
#define NN   64          // nodes
#define NH   32          // heads
#define ND   20          // edge depth
#define NE   64          // edge feat
#define KDIM 1280        // ND*NE
#define NP1  65
#define OUTP (NP1*NP1)   // 4225

#define KC      32       // K per LDS chunk (8 WMMA K-steps)
#define NCHUNK  (KDIM/KC)// 40
#define NSTEP   (KC/4)   // 8
#define ASTRIDE 36       // padded LDS row stride (floats): (4*row+col)%64 -> conflict-free
#define ROWS    256      // rows per block = 4 (b,i) pairs

#if __has_builtin(__builtin_amdgcn_global_load_async_to_lds_b128)
#define HAVE_ASYNC_LDS 1
#else
#define HAVE_ASYNC_LDS 0
#endif

// ---------------------------------------------------------------------------
// Kernel A: build Wc fragments in WMMA B-matrix order.
// Wc[kk][g] = sum_h W_edge[h, e] * edge_dis_weight[((d*32)+h)*32 + g],
//   kk = d*64 + e. Fragment layout: frag[((s*2+t)*32 + lane)*2 + u] where
//   s = K-step (k = 4s), t = N-tile: v_u = Wc[4s + 2*(lane>>4) + u][16t + (lane&15)].
// ---------------------------------------------------------------------------
__global__ __launch_bounds__(256) void build_wc_frags(
    const float* __restrict__ W_edge,   // (32,64)
    const float* __restrict__ edw,      // (128*32*32,1) -> first 20*32*32 used
    float* __restrict__ frag)           // 320*2*32*2 = 40960 floats
{
  int id = blockIdx.x * 256 + threadIdx.x;      // 0..20479
  if (id >= 320 * 2 * 32) return;
  int lane = id & 31;
  int st   = id >> 5;                           // (s*2 + t)
  int t    = st & 1;
  int g    = t * 16 + (lane & 15);
  int s    = st >> 1;
  int kk0  = 4 * s + 2 * (lane >> 4);
  float v[2];
#pragma unroll
  for (int u = 0; u < 2; ++u) {
    int kk = kk0 + u;
    int d  = kk >> 6;
    int e  = kk & 63;
    float acc = 0.f;
#pragma unroll
    for (int h = 0; h < NH; ++h)
      acc += W_edge[h * NE + e] * edw[(d * NH + h) * NH + g];
    v[u] = acc;
  }
  frag[id * 2 + 0] = v[0];
  frag[id * 2 + 1] = v[1];
}

// ---------------------------------------------------------------------------
// Kernel B: main GEMM + fused epilogue. One block per 4 (b,i) pairs: 256 rows,
// K=1280, 32 output cols. 8 wave32s; wave w owns M-tiles {2w,2w+1} x both
// N-tiles (4 v8f accumulators).
// ---------------------------------------------------------------------------
__global__ __launch_bounds__(256) void edge_gemm(
    const float* __restrict__ edge_input,   // (B,64,64,1280)
    const float* __restrict__ wfrag,        // 40960 floats, fragment order
    const float* __restrict__ attn_bias,    // (B,65,65)
    const int*   __restrict__ spatial_pos,  // (B,64,64)
    const float* __restrict__ spatial_emb,  // (513,32)
    float* __restrict__ out)                // (B,32,65,65)
{
  __shared__ __align__(16) float sA[2][ROWS * ASTRIDE];   // 2 * 36864 B

  const int tid  = threadIdx.x;
  const int lane = tid & 31;
  const int wave = tid >> 5;            // 0..7
  const int l15  = lane & 15;
  const int half = lane >> 4;

  // A copy: thread tid owns row tid of the 256-row super-block (KC floats/chunk).
  const float* Arow = edge_input + ((size_t)blockIdx.x * ROWS + tid) * KDIM;

  v8f acc[2][2];
  const v8f vzero = {0.f, 0.f, 0.f, 0.f, 0.f, 0.f, 0.f, 0.f};
  acc[0][0] = vzero; acc[0][1] = vzero; acc[1][0] = vzero; acc[1][1] = vzero;

  const int fragRow0 = (2 * wave) * 16 + l15;        // A-frag rows for this wave
  const int fragRow1 = fragRow0 + 16;

#if HAVE_ASYNC_LDS
  // ---- async direct global->LDS pipeline (no VGPR staging) ----
  const uintptr_t ldst0 = (uintptr_t)&sA[0][tid * ASTRIDE];
  const uintptr_t ldst1 = (uintptr_t)&sA[1][tid * ASTRIDE];

#define ASYNC_CHUNK(GP, LP)                                                    \
  do {                                                                         \
    as1_v4i* gv = (as1_v4i*)(uintptr_t)(GP);                                   \
    as3_v4i* lv = (as3_v4i*)(uintptr_t)(LP);                                   \
    __builtin_amdgcn_global_load_async_to_lds_b128(gv, lv, 0, 0);              \
    __builtin_amdgcn_global_load_async_to_lds_b128(gv, lv, 16, 0);             \
    __builtin_amdgcn_global_load_async_to_lds_b128(gv, lv, 32, 0);             \
    __builtin_amdgcn_global_load_async_to_lds_b128(gv, lv, 48, 0);             \
    __builtin_amdgcn_global_load_async_to_lds_b128(gv, lv, 64, 0);             \
    __builtin_amdgcn_global_load_async_to_lds_b128(gv, lv, 80, 0);             \
    __builtin_amdgcn_global_load_async_to_lds_b128(gv, lv, 96, 0);             \
    __builtin_amdgcn_global_load_async_to_lds_b128(gv, lv, 112, 0);            \
  } while (0)

  ASYNC_CHUNK(Arow, ldst0);            // prologue: chunk 0 -> buf 0
#else
  float4 ra[8];
#pragma unroll
  for (int p = 0; p < 8; ++p)
    ra[p] = *(const float4*)(Arow + p * 4);
#pragma unroll
  for (int p = 0; p < 8; ++p)
    *(float4*)(&sA[0][tid * ASTRIDE + p * 4]) = ra[p];
#endif

  for (int c = 0; c < NCHUNK; ++c) {
    const int buf = c & 1;

#if HAVE_ASYNC_LDS
    // chunk c's async LDS writes complete (async loads retire in order)
#if __has_builtin(__builtin_amdgcn_s_wait_asynccnt)
    __builtin_amdgcn_s_wait_asynccnt(0);
#else
    asm volatile("s_wait_asynccnt 0x0" ::: "memory");
#endif
    __syncthreads();                   // publish chunk c; buf^1 free of readers
    if (c + 1 < NCHUNK)                // overlaps with the WMMA loop below
      ASYNC_CHUNK(Arow + (c + 1) * KC, (buf ? ldst0 : ldst1));
#else
    __syncthreads();                   // buffer `buf` visible to all waves
    if (c + 1 < NCHUNK) {
#pragma unroll
      for (int p = 0; p < 8; ++p)
        ra[p] = *(const float4*)(Arow + (c + 1) * KC + p * 4);
    }
#endif

    const float* sAb   = &sA[buf][0];
    const float* wbase = wfrag + c * (NSTEP * 2 * 64);   // c*1024 floats

    // 8 K-steps, 4 WMMAs each (2M x 2N register blocking)
#pragma unroll
    for (int sl = 0; sl < NSTEP; ++sl) {
      // B fragments straight from L2-resident global (coalesced 256B / frag)
      const v2f b0 = *(const v2f*)(wbase + ((sl * 2 + 0) * 32 + lane) * 2);
      const v2f b1 = *(const v2f*)(wbase + ((sl * 2 + 1) * 32 + lane) * 2);
      // A fragments from LDS (conflict-free ds_load_b64)
      const v2f a0 = *(const v2f*)(&sAb[fragRow0 * ASTRIDE + 4 * sl + 2 * half]);
      const v2f a1 = *(const v2f*)(&sAb[fragRow1 * ASTRIDE + 4 * sl + 2 * half]);
      acc[0][0] = __builtin_amdgcn_wmma_f32_16x16x4_f32(false, a0, false, b0,
                                                        (short)0, acc[0][0], false, false);
      acc[0][1] = __builtin_amdgcn_wmma_f32_16x16x4_f32(false, a0, false, b1,
                                                        (short)0, acc[0][1], false, false);
      acc[1][0] = __builtin_amdgcn_wmma_f32_16x16x4_f32(false, a1, false, b0,
                                                        (short)0, acc[1][0], false, false);
      acc[1][1] = __builtin_amdgcn_wmma_f32_16x16x4_f32(false, a1, false, b1,
                                                        (short)0, acc[1][1], false, false);
    }

#if !HAVE_ASYNC_LDS
    if (c + 1 < NCHUNK) {
      const int nbuf = buf ^ 1;
#pragma unroll
      for (int p = 0; p < 8; ++p)
        *(float4*)(&sA[nbuf][tid * ASTRIDE + p * 4]) = ra[p];
    }
#endif
  }

  // ---- fused epilogue ----
  // acc[mi][t] VGPR q, lane L: row r = (2*wave+mi)*16 + q + 8*(L>>4),
  //                            col h = t*16 + (L&15)
#pragma unroll
  for (int mi = 0; mi < 2; ++mi) {
    const int mt  = 2 * wave + mi;
    const int blk = blockIdx.x * 4 + (mt >> 2);     // b*64 + i
    const int b   = blk >> 6;
    const int i   = blk & 63;
#pragma unroll
    for (int q = 0; q < 8; ++q) {
      const int r  = mt * 16 + q + 8 * half;
      const int j  = r & 63;
      const int sp = spatial_pos[(blk << 6) + j];   // b*4096 + i*64 + j
      int dv = (sp == 0) ? 1 : sp;
      dv = (dv > 1) ? dv - 1 : dv;
      dv = (dv > ND) ? ND : dv;                     // -> [1,20]
      const float rcp = 1.f / (float)dv;
      const float ab2 = 2.f * attn_bias[b * OUTP + (i + 1) * NP1 + (j + 1)];
#pragma unroll
      for (int t = 0; t < 2; ++t) {
        const int h = t * 16 + l15;
        const float spb = spatial_emb[sp * NH + h];
        out[((size_t)(b * NH + h)) * OUTP + (i + 1) * NP1 + (j + 1)] =
            ab2 + spb + acc[mi][t][q] * rcp;
      }
    }
  }
}

// ---------------------------------------------------------------------------
// Kernel C: borders. out[b,h,0,:] and out[b,h,1:,0] = 2*attn_bias + vd[h]
// ---------------------------------------------------------------------------
__global__ __launch_bounds__(160) void fill_borders(
    const float* __restrict__ attn_bias,
    const float* __restrict__ vd,      // (1,32)
    float* __restrict__ out)
{
  const int bh = blockIdx.x;           // b*32 + h
  const int b  = bh >> 5;
  const int h  = bh & 31;
  const float tt = vd[h];
  const int tid = threadIdx.x;
  if (tid < NP1) {                     // row 0, cols 0..64
    const float ab = attn_bias[b * OUTP + tid];
    out[(size_t)bh * OUTP + tid] = 2.f * ab + tt;
  } else if (tid < NP1 + NN) {         // col 0, rows 1..64
    const int r = tid - NN;            // 1..64
    const float ab = attn_bias[b * OUTP + r * NP1];
    out[(size_t)bh * OUTP + r * NP1] = 2.f * ab + tt;
  }
}

// ---------------------------------------------------------------------------
extern "C" void kernel_launch(void* const* d_in, const int* in_sizes, int n_in,
                              void* d_out, int out_size, void* d_ws, size_t ws_size,
                              hipStream_t stream) {
  (void)in_sizes; (void)n_in; (void)out_size; (void)ws_size;
  const float* attn_bias   = (const float*)d_in[1];
  const int*   spatial_pos = (const int*)d_in[2];
  const float* edge_input  = (const float*)d_in[3];
  const float* W_edge      = (const float*)d_in[5];
  const float* spatial_emb = (const float*)d_in[6];
  const float* vd          = (const float*)d_in[7];
  const float* edw         = (const float*)d_in[8];
  float* out   = (float*)d_out;
  float* wfrag = (float*)d_ws;   // 40960 floats = 160 KB

  build_wc_frags<<<80, 256, 0, stream>>>(W_edge, edw, wfrag);
  fill_borders<<<NB * NH, 160, 0, stream>>>(attn_bias, vd, out);
  edge_gemm<<<(NB * NN) / 4, 256, 0, stream>>>(edge_input, wfrag, attn_bias,
                                               spatial_pos, spatial_emb, out);
}